// Generator4DLUT_identity_32693291057271
// MI455X (gfx1250) — compile-verified
//
#include <hip/hip_runtime.h>

// Quadrilinear 4D LUT (DIM=17) on gfx1250 (MI455X).
// x:   (16, 4, 512, 512) f32   (channel plane stride = 2^18 floats)
// LUT: (4, 17^4)          f32   (channel-major)
// out: (16, 4, 512, 512) f32
//
// MI455X reasoning:
//  - streaming floor: 134MB NT traffic @ 23.3TB/s ~ 5.8us
//  - uniform-random x => zero gather coalescing; divergent L2 line requests
//    dominate. So: repack LUT into 16^4 blocks of all 16 corners x 4ch
//    (256B, 256B-aligned = exactly 2 cachelines) -> 2 lines/pixel instead
//    of 8 (entry-major) or ~32 (naive channel-major).
//  - 16.8MB block table is L2-resident (192MB L2); x/out use TH=NT so they
//    don't evict it from near caches.
//  - wave32: 256-thread blocks, 4 pixels/thread -> B128 plane accesses,
//    nibble-packed block index (shifts/ors only).

typedef float v4f __attribute__((ext_vector_type(4)));

#define LDIM   17
#define LDIM2  (17 * 17)
#define LDIM3  (17 * 17 * 17)
#define LDIM4  (17 * 17 * 17 * 17)   // 83521 entries
#define PLANE  (1 << 18)             // 512*512 floats per channel plane
#define NPIX   (16 * PLANE)          // 4,194,304 pixels
#define NGRP   (NPIX / 4)            // 1,048,576 thread-groups of 4 pixels

#define NBLK       (16 * 16 * 16 * 16)        // 65,536 base cells
#define BLK_FLOATS 64                         // 16 corners x 4 channels
#define WS_BLOCK   ((size_t)NBLK * BLK_FLOATS * sizeof(float))   // 16,777,216 B
#define WS_ENTRY   ((size_t)LDIM4 * 4 * sizeof(float))           //  1,336,336 B

// ---------------------------------------------------------------------------
// Tier-1 repack: T[b*64 + k*4 + ch] = LUT[ch][corner k of base cell b]
//   b  = (ia<<12)|(ib<<8)|(ic<<4)|il   with ia..il in [0,15]
//   k  = (da<<3)|(db<<2)|(dc<<1)|dl
// ---------------------------------------------------------------------------
__global__ __launch_bounds__(256) void lut_block_repack_kernel(
    const float* __restrict__ lut, float* __restrict__ T) {
  int tid = blockIdx.x * 256 + threadIdx.x;        // [0, NBLK*64)
  int b   = tid >> 6;
  int k   = (tid >> 2) & 15;
  int ch  = tid & 3;
  int ia = (b >> 12) & 15, ib = (b >> 8) & 15, ic = (b >> 4) & 15, il = b & 15;
  int da = (k >> 3) & 1, db = (k >> 2) & 1, dc = (k >> 1) & 1, dl = k & 1;
  int idx = (((ia + da) * LDIM + (ib + db)) * LDIM + (ic + dc)) * LDIM + (il + dl);
  T[tid] = lut[ch * LDIM4 + idx];
}

// ---------------------------------------------------------------------------
// Tier-2 repack: entry-major  T[idx*4 + c] = LUT[c*LDIM4 + idx]
// ---------------------------------------------------------------------------
__global__ __launch_bounds__(256) void lut_entry_repack_kernel(
    const float* __restrict__ lut, float* __restrict__ T) {
  int tid = blockIdx.x * 256 + threadIdx.x;
  if (tid < LDIM4 * 4) {
    int idx = tid >> 2;
    int c   = tid & 3;
    T[tid] = lut[c * LDIM4 + idx];
  }
}

// ---------------------------------------------------------------------------
// Per-axis index / fraction setup (matches reference clip/floor semantics)
// ---------------------------------------------------------------------------
__device__ __forceinline__ void axis_setup(float v, int& i, float& f) {
  float s  = fminf(fmaxf(v, 0.0f), 1.0f) * 16.0f;   // [0,16]
  float fl = fminf(floorf(s), 15.0f);               // clip(floor, 0, dim-2)
  i = (int)fl;
  f = s - fl;                                       // reaches 1.0 at v==1
}

// ---------------------------------------------------------------------------
// Tier-1 main: 256B corner blocks. 16 B128 loads off one base, 2 lines/pixel.
// ---------------------------------------------------------------------------
__global__ __launch_bounds__(256) void quadlut_block_kernel(
    const float* __restrict__ x, const float* __restrict__ T,
    float* __restrict__ out) {
  const int g    = blockIdx.x * 256 + threadIdx.x;   // group of 4 pixels
  const int p0   = g << 2;
  const int n    = p0 >> 18;
  const int rem  = p0 & (PLANE - 1);
  const int base = (n << 20) + rem;

  const v4f xa = __builtin_nontemporal_load((const v4f*)(x + base));
  const v4f xb = __builtin_nontemporal_load((const v4f*)(x + base + PLANE));
  const v4f xc = __builtin_nontemporal_load((const v4f*)(x + base + 2 * PLANE));
  const v4f xl = __builtin_nontemporal_load((const v4f*)(x + base + 3 * PLANE));

  v4f r0, r1, r2, r3;  // per-channel results across the 4 pixels

#pragma unroll
  for (int j = 0; j < 4; ++j) {
    int ia, ib, ic, il;
    float fa, fb, fc, fl;
    axis_setup(xa[j], ia, fa);
    axis_setup(xb[j], ib, fb);
    axis_setup(xc[j], ic, fc);
    axis_setup(xl[j], il, fl);

    // Nibble-packed block index: shifts/ors only.
    const int bidx = (ia << 12) | (ib << 8) | (ic << 4) | il;
    const v4f* __restrict__ B = (const v4f*)(T + (bidx << 6));

    const float wa0 = 1.0f - fa, wb0 = 1.0f - fb;
    const float wc0 = 1.0f - fc, wl0 = 1.0f - fl;
    const float wab[4] = {wa0 * wb0, wa0 * fb, fa * wb0, fa * fb};   // (da<<1)|db
    const float wcl[4] = {wc0 * wl0, wc0 * fl, fc * wl0, fc * fl};   // (dc<<1)|dl

    v4f acc = {0.0f, 0.0f, 0.0f, 0.0f};
#pragma unroll
    for (int k = 0; k < 16; ++k) {           // k = (da<<3)|(db<<2)|(dc<<1)|dl
      const float w = wab[k >> 2] * wcl[k & 3];
      acc += w * B[k];                       // immediate-offset B128 loads
    }

    r0[j] = acc.x; r1[j] = acc.y; r2[j] = acc.z; r3[j] = acc.w;
  }

  __builtin_nontemporal_store(r0, (v4f*)(out + base));
  __builtin_nontemporal_store(r1, (v4f*)(out + base + PLANE));
  __builtin_nontemporal_store(r2, (v4f*)(out + base + 2 * PLANE));
  __builtin_nontemporal_store(r3, (v4f*)(out + base + 3 * PLANE));
}

// abc corner offsets (in LUT entries) for tiers 2/3; l handled by paired load.
__device__ __constant__ int kCornerOff[8] = {
    0,             LDIM,
    LDIM2,         LDIM2 + LDIM,
    LDIM3,         LDIM3 + LDIM,
    LDIM3 + LDIM2, LDIM3 + LDIM2 + LDIM};

// ---------------------------------------------------------------------------
// Tier-2 main: entry-major LUT (two B128 per abc corner), 8 lines/pixel.
// ---------------------------------------------------------------------------
__global__ __launch_bounds__(256) void quadlut_entry_kernel(
    const float* __restrict__ x, const float* __restrict__ T,
    float* __restrict__ out) {
  const int g    = blockIdx.x * 256 + threadIdx.x;
  const int p0   = g << 2;
  const int n    = p0 >> 18;
  const int rem  = p0 & (PLANE - 1);
  const int base = (n << 20) + rem;

  const v4f xa = __builtin_nontemporal_load((const v4f*)(x + base));
  const v4f xb = __builtin_nontemporal_load((const v4f*)(x + base + PLANE));
  const v4f xc = __builtin_nontemporal_load((const v4f*)(x + base + 2 * PLANE));
  const v4f xl = __builtin_nontemporal_load((const v4f*)(x + base + 3 * PLANE));

  v4f r0, r1, r2, r3;

#pragma unroll
  for (int j = 0; j < 4; ++j) {
    int ia, ib, ic, il;
    float fa, fb, fc, fl;
    axis_setup(xa[j], ia, fa);
    axis_setup(xb[j], ib, fb);
    axis_setup(xc[j], ic, fc);
    axis_setup(xl[j], il, fl);

    const int ibase = ((ia * LDIM + ib) * LDIM + ic) * LDIM + il;
    const float* __restrict__ Tb = T + (ibase << 2);

    const float wa0 = 1.0f - fa, wb0 = 1.0f - fb, wc0 = 1.0f - fc;
    const float wab[4] = {wa0 * wb0, wa0 * fb, fa * wb0, fa * fb};

    v4f acc = {0.0f, 0.0f, 0.0f, 0.0f};
#pragma unroll
    for (int k = 0; k < 8; ++k) {
      const float w = wab[k >> 1] * ((k & 1) ? fc : wc0);
      const float* p = Tb + (kCornerOff[k] << 2);
      v4f v0 = *(const v4f*)(p);
      v4f v1 = *(const v4f*)(p + 4);
      v4f vl = v0 + fl * (v1 - v0);
      acc += w * vl;
    }

    r0[j] = acc.x; r1[j] = acc.y; r2[j] = acc.z; r3[j] = acc.w;
  }

  __builtin_nontemporal_store(r0, (v4f*)(out + base));
  __builtin_nontemporal_store(r1, (v4f*)(out + base + PLANE));
  __builtin_nontemporal_store(r2, (v4f*)(out + base + 2 * PLANE));
  __builtin_nontemporal_store(r3, (v4f*)(out + base + 3 * PLANE));
}

// ---------------------------------------------------------------------------
// Tier-3 main (no usable ws): original channel-major LUT, scalar l-pairs.
// ---------------------------------------------------------------------------
__global__ __launch_bounds__(256) void quadlut_fallback_kernel(
    const float* __restrict__ x, const float* __restrict__ lut,
    float* __restrict__ out) {
  const int g    = blockIdx.x * 256 + threadIdx.x;
  const int p0   = g << 2;
  const int n    = p0 >> 18;
  const int rem  = p0 & (PLANE - 1);
  const int base = (n << 20) + rem;

  const v4f xa = __builtin_nontemporal_load((const v4f*)(x + base));
  const v4f xb = __builtin_nontemporal_load((const v4f*)(x + base + PLANE));
  const v4f xc = __builtin_nontemporal_load((const v4f*)(x + base + 2 * PLANE));
  const v4f xl = __builtin_nontemporal_load((const v4f*)(x + base + 3 * PLANE));

  v4f r0, r1, r2, r3;

#pragma unroll
  for (int j = 0; j < 4; ++j) {
    int ia, ib, ic, il;
    float fa, fb, fc, fl;
    axis_setup(xa[j], ia, fa);
    axis_setup(xb[j], ib, fb);
    axis_setup(xc[j], ic, fc);
    axis_setup(xl[j], il, fl);

    const int ibase = ((ia * LDIM + ib) * LDIM + ic) * LDIM + il;
    const float wa0 = 1.0f - fa, wb0 = 1.0f - fb, wc0 = 1.0f - fc;
    const float wab[4] = {wa0 * wb0, wa0 * fb, fa * wb0, fa * fb};

    float a0 = 0.f, a1 = 0.f, a2 = 0.f, a3 = 0.f;
#pragma unroll
    for (int k = 0; k < 8; ++k) {
      const float w   = wab[k >> 1] * ((k & 1) ? fc : wc0);
      const int   idx = ibase + kCornerOff[k];
#pragma unroll
      for (int ch = 0; ch < 4; ++ch) {
        const float* p = lut + ch * LDIM4 + idx;
        float vl = p[0] + fl * (p[1] - p[0]);
        if (ch == 0) a0 += w * vl;
        else if (ch == 1) a1 += w * vl;
        else if (ch == 2) a2 += w * vl;
        else a3 += w * vl;
      }
    }
    r0[j] = a0; r1[j] = a1; r2[j] = a2; r3[j] = a3;
  }

  __builtin_nontemporal_store(r0, (v4f*)(out + base));
  __builtin_nontemporal_store(r1, (v4f*)(out + base + PLANE));
  __builtin_nontemporal_store(r2, (v4f*)(out + base + 2 * PLANE));
  __builtin_nontemporal_store(r3, (v4f*)(out + base + 3 * PLANE));
}

// ---------------------------------------------------------------------------
extern "C" void kernel_launch(void* const* d_in, const int* in_sizes, int n_in,
                              void* d_out, int out_size, void* d_ws,
                              size_t ws_size, hipStream_t stream) {
  const float* x   = (const float*)d_in[0];
  const float* lut = (const float*)d_in[1];
  float*       out = (float*)d_out;

  if (d_ws != nullptr && ws_size >= WS_BLOCK) {
    // Tier 1: 256B corner blocks -> 2 cachelines/pixel of gather.
    float* T = (float*)d_ws;
    lut_block_repack_kernel<<<(NBLK * BLK_FLOATS) / 256, 256, 0, stream>>>(lut, T);
    quadlut_block_kernel<<<NGRP / 256, 256, 0, stream>>>(x, T, out);
  } else if (d_ws != nullptr && ws_size >= WS_ENTRY) {
    // Tier 2: entry-major repack -> 8 cachelines/pixel.
    float* T = (float*)d_ws;
    lut_entry_repack_kernel<<<(LDIM4 * 4 + 255) / 256, 256, 0, stream>>>(lut, T);
    quadlut_entry_kernel<<<NGRP / 256, 256, 0, stream>>>(x, T, out);
  } else {
    // Tier 3: direct channel-major gathers.
    quadlut_fallback_kernel<<<NGRP / 256, 256, 0, stream>>>(x, lut, out);
  }
}